// InfoNCE_Loss_90649579749419
// MI455X (gfx1250) — compile-verified
//
#include <hip/hip_runtime.h>
#include <hip/hip_bf16.h>

typedef __attribute__((ext_vector_type(2))) float    v2f;
typedef __attribute__((ext_vector_type(4))) float    v4f;
typedef __attribute__((ext_vector_type(8))) float    v8f;
typedef __attribute__((ext_vector_type(4))) unsigned v4u;
typedef __attribute__((ext_vector_type(4))) int      v4i;
typedef __attribute__((ext_vector_type(8))) int      v8i;

#define PRED_STEPS 5
#define NEG  16
#define Cc   256
#define Bb   64
#define Hh   16
#define Ww   16
#define SKIP 1
#define EPSV 1e-11f

// ---------------------------------------------------------------------------
// Kernel 1: ztwk[r][o] = sum_i Wk[o,i] * z[b,i,h_rel+off,w],  r=((h_rel*W+w)*B+b)
// Block: 512 threads = 16 waves; tile = fixed (h_rel,b), w = 0..15 so the
// 16 z-columns form a dense 2D tile: 256 channel-rows x 64B, row stride 1KB.
// Wave 0 issues one TENSOR_LOAD_TO_LDS (TDM) for the whole 16KB stage.
// Each wave then computes a 16(M)x16(N) tile with V_WMMA_F32_16X16X4_F32.
// ---------------------------------------------------------------------------
__global__ __launch_bounds__(512) void cpc_proj_gemm(
    const float* __restrict__ z, const float* __restrict__ Wk_all,
    float* __restrict__ ztwk, int k)
{
    const int off = k + SKIP;
    const int hb  = blockIdx.x;                       // [0, Hk*B)
    const int hr  = hb / Bb;                          // h_rel
    const int b   = hb % Bb;

    const float* Wmat = Wk_all + (size_t)(k - 1) * Cc * Cc;

    __shared__ float lds_z[Cc][16];                   // [channel][w], 16 KB

    const int t    = threadIdx.x;
    const int lane = t & 31;
    const int wave = t >> 5;                          // 0..15

    // ---- TDM stage: one descriptor, issued by wave 0 only (uniform branch).
    if (wave == 0) {
        const uint64_t ga =
            (uint64_t)(z + ((size_t)b * Cc * Hh + (hr + off)) * Ww); // i=0,w=0
        const unsigned lds_off = (unsigned)(size_t)(void*)&lds_z[0][0];

        // D# group 0: count=1 | lds_addr | global_addr[56:0] | type=2
        v4u g0;
        g0[0] = 1u;                                   // count=1, user mode
        g0[1] = lds_off;                              // LDS byte address
        g0[2] = (unsigned)ga;                         // global_addr[31:0]
        g0[3] = (unsigned)((ga >> 32) & 0x1FFFFFFu) | (2u << 30); // [56:32]|type

        // D# group 1: data_size=4B; tensor 16 x 256, tile 16 x 256,
        // dim0 stride = H*W = 256 elements.
        v8i g1;
        g1[0] = (int)(2u << 16);                      // wg_mask=0, data_size=2 (4B)
        g1[1] = (int)((unsigned)(Ww & 0xFFFF) << 16); // tensor_dim0[15:0] @63:48
        g1[2] = (int)((unsigned)(Cc & 0xFFFF) << 16); // dim0[31:16]=0 | tensor_dim1[15:0]
        g1[3] = (int)((unsigned)Ww << 16);            // tensor_dim1[31:16]=0 | tile_dim0
        g1[4] = (int)Cc;                              // tile_dim1=256, tile_dim2=0
        g1[5] = (int)(Hh * Ww);                       // tensor_dim0_stride[31:0]
        g1[6] = 0;                                    // stride[47:32] | dim1_stride lo
        g1[7] = 0;

        v4i gz4 = {0, 0, 0, 0};                       // 2D tensor: groups 2/3 unused
        v8i gz8 = {0, 0, 0, 0, 0, 0, 0, 0};
        __builtin_amdgcn_tensor_load_to_lds(g0, g1, gz4, gz4, gz8, 0);
        __builtin_amdgcn_s_wait_tensorcnt(0);
    }
    __syncthreads();

    // ---- WMMA K-loop.  fp32 16x16x4 lane layout:
    //   A: lanes 0-15 -> M=lane,    K={i0,i0+1}; lanes 16-31 -> M=lane-16, K={i0+2,i0+3}
    //   B: lanes 0-15 -> N=lane,    K={i0,i0+1}; lanes 16-31 -> N=lane-16, K={i0+2,i0+3}
    const int m0    = wave * 16;
    const int mlane = m0 + (lane & 15);
    const int ksel  = (lane >> 4) * 2;                // 0 or 2
    const int col   = lane & 15;                      // = w
    const float* Arow = Wmat + (size_t)mlane * Cc + ksel;
    __builtin_prefetch(Arow, 0, 0);                   // global_prefetch_b8

    v8f acc0 = {};
    v8f acc1 = {};
#pragma unroll 8
    for (int i0 = 0; i0 < Cc; i0 += 8) {
        v2f a0 = *(const v2f*)(Arow + i0);
        v2f b0;
        b0[0] = lds_z[i0 + ksel + 0][col];
        b0[1] = lds_z[i0 + ksel + 1][col];
        acc0 = __builtin_amdgcn_wmma_f32_16x16x4_f32(
            false, a0, false, b0, (short)0, acc0, false, false);
        v2f a1 = *(const v2f*)(Arow + i0 + 4);
        v2f b1;
        b1[0] = lds_z[i0 + 4 + ksel + 0][col];
        b1[1] = lds_z[i0 + 4 + ksel + 1][col];
        acc1 = __builtin_amdgcn_wmma_f32_16x16x4_f32(
            false, a1, false, b1, (short)0, acc1, false, false);
    }
    v8f acc = acc0 + acc1;

    // ---- store D: lane holds 8 contiguous output channels of one column
    const int r = ((hr * Ww + col) * Bb) + b;         // reference flat row index
    float* orow = ztwk + (size_t)r * Cc + m0 + ((lane >> 4) * 8);
    v4f lo = {acc[0], acc[1], acc[2], acc[3]};
    v4f hi = {acc[4], acc[5], acc[6], acc[7]};
    *(v4f*)(orow + 0) = lo;
    *(v4f*)(orow + 4) = hi;
}

// ---------------------------------------------------------------------------
// Kernel 2: per position r -> 17 dot products (main + 16 gathered negatives),
// softmax over 17, loss = -log(p0 + eps); scatter into patch map + scalar.
// One wave per r; lane owns 8 channels; ztwk rows are contiguous 1KB.
// ---------------------------------------------------------------------------
__global__ __launch_bounds__(256) void cpc_score(
    const float* __restrict__ cctx, const float* __restrict__ ztwk,
    const int* __restrict__ rand_idx, float* __restrict__ patch_loss,
    float* __restrict__ total_acc, int k)
{
    const int off  = k + SKIP;
    const int Hk   = Hh - off;
    const int Nk   = Hk * Ww * Bb;
    const int wave = threadIdx.x >> 5;
    const int lane = threadIdx.x & 31;
    const int r    = blockIdx.x * 8 + wave;
    if (r >= Nk) return;

    const int b  = r % Bb;
    const int hw = r / Bb;
    const int w_ = hw % Ww;
    const int hr = hw / Ww;                           // h_rel in [0,Hk)

    const float* ccol = cctx + (((size_t)b * Cc) * Hh + hr) * Ww + w_;
    float ctx[8];
#pragma unroll
    for (int u = 0; u < 8; ++u)
        ctx[u] = ccol[(size_t)(lane * 8 + u) * Hh * Ww];

    float logit[NEG + 1];
#pragma unroll 1
    for (int tg = 0; tg <= NEG; ++tg) {
        const int row = (tg == 0) ? r : rand_idx[(size_t)r * NEG + (tg - 1)];
        const float* zr = ztwk + (size_t)row * Cc + lane * 8;
        v4f p0 = *(const v4f*)(zr);
        v4f p1 = *(const v4f*)(zr + 4);
        float d = ctx[0]*p0[0] + ctx[1]*p0[1] + ctx[2]*p0[2] + ctx[3]*p0[3]
                + ctx[4]*p1[0] + ctx[5]*p1[1] + ctx[6]*p1[2] + ctx[7]*p1[3];
#pragma unroll
        for (int s = 16; s >= 1; s >>= 1) d += __shfl_xor(d, s, 32);
        logit[tg] = d;
    }

    float m = logit[0];
#pragma unroll
    for (int tg = 1; tg <= NEG; ++tg) m = fmaxf(m, logit[tg]);
    float sum = 0.f;
#pragma unroll
    for (int tg = 0; tg <= NEG; ++tg) sum += __expf(logit[tg] - m);
    const float p0   = __expf(logit[0] - m) / sum;
    const float loss = -__logf(p0 + EPSV);

    if (lane == 0) {
        atomicAdd(&patch_loss[((size_t)b * Hh + (hr + off)) * Ww + w_], loss);
        atomicAdd(&patch_loss[((size_t)b * Hh + hr) * Ww + w_], loss);
        atomicAdd(total_acc, loss * (1.0f / ((float)PRED_STEPS * (float)Nk)));
    }
}

// ---------------------------------------------------------------------------
__global__ void cpc_init(float* __restrict__ out, float* __restrict__ total_acc)
{
    const int i = blockIdx.x * 256 + threadIdx.x;
    if (i < 1 + Bb * Hh * Ww) out[i] = 0.f;           // total + patchwise_loss
    if (i == 0) *total_acc = 0.f;
}

__global__ void cpc_finalize(const float* __restrict__ total_acc,
                             float* __restrict__ out_total,
                             float* __restrict__ out_count)
{
    const int t = threadIdx.x;                        // 256 = H*W
    if (t == 0) *out_total = *total_acc;
    const int h = t / Ww;
    float cnt = 0.f;
#pragma unroll
    for (int k = 1; k <= PRED_STEPS; ++k) {
        const int o = k + SKIP;
        if (h >= o)      cnt += 1.f;
        if (h < Hh - o)  cnt += 1.f;
    }
    out_count[t] = cnt;
}

// ---------------------------------------------------------------------------
extern "C" void kernel_launch(void* const* d_in, const int* in_sizes, int n_in,
                              void* d_out, int out_size, void* d_ws, size_t ws_size,
                              hipStream_t stream)
{
    const float* z  = (const float*)d_in[0];
    const float* c  = (const float*)d_in[1];
    const float* Wk = (const float*)d_in[2];
    const int* ridx[PRED_STEPS] = {
        (const int*)d_in[3], (const int*)d_in[4], (const int*)d_in[5],
        (const int*)d_in[6], (const int*)d_in[7]};

    float* out       = (float*)d_out;                 // [0]=total, [1..]=patch, tail=count
    float* total_acc = (float*)d_ws;                  // 1 float
    float* ztwk      = (float*)((char*)d_ws + 256);   // max 14336*256 floats, reused per k

    cpc_init<<<(1 + Bb * Hh * Ww + 255) / 256, 256, 0, stream>>>(out, total_acc);

    for (int k = 1; k <= PRED_STEPS; ++k) {
        const int Hk = Hh - (k + SKIP);
        cpc_proj_gemm<<<dim3(Hk * Bb), 512, 0, stream>>>(z, Wk, ztwk, k);
        const int Nk = Hk * Ww * Bb;
        cpc_score<<<dim3((Nk + 7) / 8), 256, 0, stream>>>(
            c, ztwk, ridx[k - 1], out + 1, total_acc, k);
    }

    cpc_finalize<<<1, 256, 0, stream>>>(total_acc, out, out + 1 + Bb * Hh * Ww);
}